// IncrementalSpatialAttention_53446573031913
// MI455X (gfx1250) — compile-verified
//
#include <hip/hip_runtime.h>
#include <hip/hip_bf16.h>
#include <math.h>

typedef __attribute__((ext_vector_type(2))) float v2f;
typedef __attribute__((ext_vector_type(8))) float v8f;

#define HEADS   15
#define DHEAD   64
#define DIM     960
#define INNER   960
#define QKVW    (3 * INNER)        // 2880
#define BATCH   32
#define SEQ     1024
#define TMAX    61                 // max token_limit = 1 + 15*4
#define MROWS   (BATCH * TMAX)     // 1952 (divisible by 32)
#define SUMTL   495                // 5+9+...+61
#define OROWS   (BATCH * SUMTL)    // 15840 (divisible by 32)

// ---------------------------------------------------------------------------
// K0: gather the first 61 tokens of each batch into a compact [1952, 960]
// matrix so the GEMM A-side addressing is dense.
// ---------------------------------------------------------------------------
__global__ void gather_x_kernel(const float* __restrict__ x, float* __restrict__ xg) {
    int idx = blockIdx.x * 256 + threadIdx.x;
    if (idx >= MROWS * DIM) return;
    int r = idx / DIM;
    int c = idx - r * DIM;
    int b = r / TMAX;
    int t = r - b * TMAX;
    xg[idx] = x[((size_t)b * SEQ + t) * DIM + c];
}

// ---------------------------------------------------------------------------
// Tiled transpose: out[N,K] = in[K,N]^T. K,N multiples of 32.
// Makes every GEMM B-fragment a contiguous 8-byte load.
// ---------------------------------------------------------------------------
__global__ void transpose_kernel(const float* __restrict__ in, float* __restrict__ out,
                                 int K, int N) {
    __shared__ float tile[32][33];
    const int n0 = blockIdx.x * 32;
    const int k0 = blockIdx.y * 32;
    const int tx = threadIdx.x;            // 0..31
    const int ty = threadIdx.y;            // 0..7
    for (int i = ty; i < 32; i += 8) {
        tile[i][tx] = in[(size_t)(k0 + i) * N + n0 + tx];
    }
    __syncthreads();
    for (int i = ty; i < 32; i += 8) {
        out[(size_t)(n0 + i) * K + k0 + tx] = tile[tx][i];
    }
}

// ---------------------------------------------------------------------------
// Fragment bundle for one K-step: 2 A-frags (2 M-tiles) + 4 B-frags (4 N-tiles)
// ---------------------------------------------------------------------------
struct Frags {
    v2f a0, a1, b0, b1, b2, b3;
};

__device__ __forceinline__ Frags load_frags(const float* pa0, const float* pa1,
                                            const float* pb0, const float* pb1,
                                            const float* pb2, const float* pb3,
                                            int koff) {
    Frags f;
    // A first: each A fragment feeds 4 WMMAs (critical operands).
    f.a0 = *(const v2f*)(pa0 + koff);
    f.a1 = *(const v2f*)(pa1 + koff);
    f.b0 = *(const v2f*)(pb0 + koff);
    f.b1 = *(const v2f*)(pb1 + koff);
    f.b2 = *(const v2f*)(pb2 + koff);
    f.b3 = *(const v2f*)(pb3 + koff);
    return f;
}

__device__ __forceinline__ void wmma8(const Frags& f, v8f acc[8]) {
    acc[0] = __builtin_amdgcn_wmma_f32_16x16x4_f32(false, f.a0, false, f.b0, (short)0, acc[0], false, false);
    acc[1] = __builtin_amdgcn_wmma_f32_16x16x4_f32(false, f.a0, false, f.b1, (short)0, acc[1], false, false);
    acc[2] = __builtin_amdgcn_wmma_f32_16x16x4_f32(false, f.a0, false, f.b2, (short)0, acc[2], false, false);
    acc[3] = __builtin_amdgcn_wmma_f32_16x16x4_f32(false, f.a0, false, f.b3, (short)0, acc[3], false, false);
    acc[4] = __builtin_amdgcn_wmma_f32_16x16x4_f32(false, f.a1, false, f.b0, (short)0, acc[4], false, false);
    acc[5] = __builtin_amdgcn_wmma_f32_16x16x4_f32(false, f.a1, false, f.b1, (short)0, acc[5], false, false);
    acc[6] = __builtin_amdgcn_wmma_f32_16x16x4_f32(false, f.a1, false, f.b2, (short)0, acc[6], false, false);
    acc[7] = __builtin_amdgcn_wmma_f32_16x16x4_f32(false, f.a1, false, f.b3, (short)0, acc[7], false, false);
}

// ---------------------------------------------------------------------------
// WMMA fp32 GEMM: C[M,N] = A[M,K] @ B[K,N] (+ bias[N] if non-null).
// B supplied TRANSPOSED (BT[N,K]) -> every fragment is a contiguous b64 load.
// One wave per block computes a 32-row x 64-col strip (2 M-tiles x 4 N-tiles,
// 8 WMMAs per K-step). Prefetch distance 2 (two fragment stages): loads for
// step k+8 are issued before the WMMAs of step k, giving each load ~16 WMMAs
// plus a load-group of cover before its first consumer.
// Requires: K multiple of 8, M of 32, N of 64.
//
// Fragment layouts (ISA 7.12.2, wave32):
//   A 16x4 : lane L: v0 = A[m0+L%16][k+2*(L/16)], v1 = A[..][k+2*(L/16)+1]
//   B 4x16 : lane L: v0 = BT[n0+L%16][k+2*(L/16)], v1 = BT[..][k+2*(L/16)+1]
//   C/D    : vgpr r: lanes 0-15 -> M=m0+r, lanes 16-31 -> M=m0+8+r; N=n0+L%16
// ---------------------------------------------------------------------------
__global__ void gemm_wmma_f32_kernel(const float* __restrict__ A,
                                     const float* __restrict__ BT,
                                     float* __restrict__ C,
                                     const float* __restrict__ bias,
                                     int M, int N, int K) {
    const int lane = threadIdx.x & 31;
    const int lo = lane & 15;
    const int hi = lane >> 4;              // 0 or 1
    const int m0 = blockIdx.y * 32;
    const int n0 = blockIdx.x * 64;
    const int kaOff = 2 * hi;

    // Per-lane streaming pointers (rows are 8B-aligned: K is even).
    const float* __restrict__ pa0 = A  + (size_t)(m0 + lo) * K + kaOff;
    const float* __restrict__ pa1 = A  + (size_t)(m0 + 16 + lo) * K + kaOff;
    const float* __restrict__ pb0 = BT + (size_t)(n0 +  0 + lo) * K + kaOff;
    const float* __restrict__ pb1 = BT + (size_t)(n0 + 16 + lo) * K + kaOff;
    const float* __restrict__ pb2 = BT + (size_t)(n0 + 32 + lo) * K + kaOff;
    const float* __restrict__ pb3 = BT + (size_t)(n0 + 48 + lo) * K + kaOff;

    v8f acc[8];
#pragma unroll
    for (int t = 0; t < 8; ++t) acc[t] = (v8f){};

    // Prologue: stages for k = 0 and k = 4.
    Frags f0 = load_frags(pa0, pa1, pb0, pb1, pb2, pb3, 0);
    Frags f1 = load_frags(pa0, pa1, pb0, pb1, pb2, pb3, 4);

    // Main loop: k, k+4 computed; k+8, k+12 prefetched.  (K mult of 8)
    for (int k = 0; k < K - 8; k += 8) {
        Frags f0n = load_frags(pa0, pa1, pb0, pb1, pb2, pb3, 8);
        wmma8(f0, acc);
        Frags f1n = load_frags(pa0, pa1, pb0, pb1, pb2, pb3, 12);
        wmma8(f1, acc);
        f0 = f0n;
        f1 = f1n;
        pa0 += 8; pa1 += 8; pb0 += 8; pb1 += 8; pb2 += 8; pb3 += 8;
    }
    // Epilogue: last two K-steps (no prefetch).
    wmma8(f0, acc);
    wmma8(f1, acc);

    // Store: m-tile mt rows m0 + 16*mt + 8*hi + r, col n0 + 16*t + lo.
#pragma unroll
    for (int mt = 0; mt < 2; ++mt) {
        float* __restrict__ crow = C + (size_t)(m0 + 16 * mt + 8 * hi) * N + n0 + lo;
#pragma unroll
        for (int t = 0; t < 4; ++t) {
            float bv = bias ? bias[n0 + t * 16 + lo] : 0.0f;
#pragma unroll
            for (int r = 0; r < 8; ++r) {
                crow[(size_t)r * N + t * 16] = acc[4 * mt + t][r] + bv;
            }
        }
    }
}

// ---------------------------------------------------------------------------
// K2: attention for one (head, batch, window) triple per block (64 threads).
// Windows w=0..14 have tl = 5+4w keys/queries. K/V tiles (<=61x64 f32) are
// staged in LDS; exact softmax; output row layout matches the concatenated
// tuple order of the reference: row = 32*(2w^2+3w) + b*tl + i, col = h*64+d.
// ---------------------------------------------------------------------------
__global__ void attn_kernel(const float* __restrict__ qkv, float* __restrict__ O) {
    __shared__ float Ks[TMAX * DHEAD];
    __shared__ float Vs[TMAX * DHEAD];
    __shared__ float qv[DHEAD];
    __shared__ float sc[64];
    __shared__ float red[2];

    const int h = blockIdx.x;
    const int b = blockIdx.y;
    const int w = blockIdx.z;
    const int tl = 5 + 4 * w;
    const int tid = threadIdx.x;           // 0..63
    const int rb = b * TMAX;

    const float* __restrict__ qbase = qkv + (size_t)rb * QKVW + h * DHEAD;
    const float* __restrict__ kbase = qbase + INNER;
    const float* __restrict__ vbase = qbase + 2 * INNER;

    for (int idx = tid; idx < tl * DHEAD; idx += 64) {
        int j = idx >> 6;
        int d = idx & 63;
        Ks[idx] = kbase[(size_t)j * QKVW + d];
        Vs[idx] = vbase[(size_t)j * QKVW + d];
    }
    __syncthreads();

    const int rowOff = 32 * (2 * w * w + 3 * w);   // 32 * sum_{w'<w} tl_{w'}
    const float scale = 0.125f;                     // 64^-0.5

    for (int i = 0; i < tl; ++i) {
        qv[tid] = qbase[(size_t)i * QKVW + tid];
        __syncthreads();

        if (tid < tl) {
            float s = 0.0f;
#pragma unroll 8
            for (int d = 0; d < DHEAD; ++d) s += qv[d] * Ks[tid * DHEAD + d];
            sc[tid] = s * scale;
        }
        __syncthreads();

        if (tid == 0) {
            float m = sc[0];
            for (int j = 1; j < tl; ++j) m = fmaxf(m, sc[j]);
            red[0] = m;
        }
        __syncthreads();

        if (tid < tl) sc[tid] = expf(sc[tid] - red[0]);
        __syncthreads();

        if (tid == 0) {
            float s = 0.0f;
            for (int j = 0; j < tl; ++j) s += sc[j];
            red[1] = s;
        }
        __syncthreads();

        float o = 0.0f;
        for (int j = 0; j < tl; ++j) o += sc[j] * Vs[j * DHEAD + tid];
        O[(size_t)(rowOff + b * tl + i) * DIM + h * DHEAD + tid] = o / red[1];
        __syncthreads();
    }
}

// ---------------------------------------------------------------------------
// Launch
// ---------------------------------------------------------------------------
extern "C" void kernel_launch(void* const* d_in, const int* in_sizes, int n_in,
                              void* d_out, int out_size, void* d_ws, size_t ws_size,
                              hipStream_t stream) {
    const float* x    = (const float*)d_in[0];   // [32, 1024, 960]
    const float* Wqkv = (const float*)d_in[1];   // [960, 2880]
    const float* Wout = (const float*)d_in[2];   // [960, 960]
    const float* bout = (const float*)d_in[3];   // [960]
    float* out = (float*)d_out;                  // 32*495*960 floats, concat order

    float* xg    = (float*)d_ws;                           // [1952, 960]
    float* qkv   = xg    + (size_t)MROWS * DIM;            // [1952, 2880]
    float* O     = qkv   + (size_t)MROWS * QKVW;           // [15840, 960]
    float* WqkvT = O     + (size_t)OROWS * DIM;            // [2880, 960]
    float* WoutT = WqkvT + (size_t)QKVW * DIM;             // [960, 960]

    // K0: compact first-61-token rows
    gather_x_kernel<<<(MROWS * DIM + 255) / 256, 256, 0, stream>>>(x, xg);

    // One-time weight transposes (contiguous-K B fragments for the GEMMs).
    transpose_kernel<<<dim3(QKVW / 32, DIM / 32), dim3(32, 8), 0, stream>>>(Wqkv, WqkvT, DIM, QKVW);
    transpose_kernel<<<dim3(DIM / 32, INNER / 32), dim3(32, 8), 0, stream>>>(Wout, WoutT, INNER, DIM);

    // K1: qkv = xg @ Wqkv     (M=1952, N=2880, K=960)
    gemm_wmma_f32_kernel<<<dim3(QKVW / 64, MROWS / 32), 32, 0, stream>>>(
        xg, WqkvT, qkv, nullptr, MROWS, QKVW, DIM);

    // K2: all 15 windows x 15 heads x 32 batches of attention
    attn_kernel<<<dim3(HEADS, BATCH, 15), 64, 0, stream>>>(qkv, O);

    // K3: out = O @ Wout + bout   (M=15840, N=960, K=960), writes d_out directly
    gemm_wmma_f32_kernel<<<dim3(DIM / 64, OROWS / 32), 32, 0, stream>>>(
        O, WoutT, out, bout, OROWS, DIM, INNER);
}